// VideoTubeExtractor_46188078301434
// MI455X (gfx1250) — compile-verified
//
#include <hip/hip_runtime.h>
#include <hip/hip_bf16.h>
#include <stdint.h>

// Problem constants (from the reference)
#define TUBE      32
#define NTUBES    8
#define NB        8
#define NF        16
#define NC        3
#define NH        224
#define NW        224

typedef uint32_t u32x4 __attribute__((ext_vector_type(4)));
typedef int32_t  i32x8 __attribute__((ext_vector_type(8)));
typedef int32_t  i32x4 __attribute__((ext_vector_type(4)));

// One workgroup (one wave32) per (b, tube, f). It DMA-copies a
// 32(x) x 32(y) x 3(c) strided tile from the video into LDS with the
// Tensor Data Mover, waits on TENSORcnt, then DMA-stores the contiguous
// 12KB tile to the output. No VGPR data staging at all.
__global__ __launch_bounds__(32)
void VideoTube_tdm_kernel(const float* __restrict__ video,
                          const int*   __restrict__ yoff,
                          const int*   __restrict__ xoff,
                          float*       __restrict__ out)
{
    __shared__ float tile[NC * TUBE * TUBE];   // 12 KB per workgroup

    const int wg = blockIdx.x;                 // 0..1023 == (b*NTUBES + t)*NF + f
    const int f  = wg % NF;
    const int bt = wg / NF;
    const int t  = bt % NTUBES;
    const int b  = bt / NTUBES;

    // Per-workgroup uniform crop offsets -> force to SGPRs
    int y = __builtin_amdgcn_readfirstlane(yoff[b * NTUBES + t]);
    int x = __builtin_amdgcn_readfirstlane(xoff[b * NTUBES + t]);

    const uint64_t plane    = (uint64_t)NH * NW;               // 50176 elements
    const uint64_t src_elem = ((uint64_t)b * NF + f) * (NC * plane)
                            + (uint64_t)y * NW + (uint64_t)x;  // tile start (c=0)
    const uint64_t dst_elem = (uint64_t)wg * (NC * TUBE * TUBE);

    const uint64_t src_addr = (uint64_t)(uintptr_t)video + src_elem * 4u;
    const uint64_t dst_addr = (uint64_t)(uintptr_t)out   + dst_elem * 4u;
    // Low 32 bits of a flat LDS pointer == byte offset in the group segment
    const uint32_t lds_addr = (uint32_t)(uintptr_t)(&tile[0]);

#if defined(__gfx1250__) && __has_builtin(__builtin_amdgcn_tensor_load_to_lds) && \
    __has_builtin(__builtin_amdgcn_tensor_store_from_lds)
    // ---------------- TDM load descriptor: 3D tile (x=32, y=32, c=3) ----------
    u32x4 ld_g0;
    ld_g0[0] = 1u;                                             // count=1 (valid), user mode
    ld_g0[1] = lds_addr;                                       // LDS dest (bytes)
    ld_g0[2] = (uint32_t)(src_addr & 0xFFFFFFFFu);             // global_addr[31:0]
    ld_g0[3] = (uint32_t)((src_addr >> 32) & 0x01FFFFFFu)      // global_addr[56:32]
             | (2u << 30);                                     // type = 2 ("image")

    i32x8 ld_g1;
    ld_g1[0] = (int32_t)(2u << 16);                            // data_size = 4 bytes
    ld_g1[1] = (int32_t)(((uint32_t)NW & 0xFFFFu) << 16);      // tensor_dim0 lo16 (=224)
    ld_g1[2] = (int32_t)((((uint32_t)NW) >> 16)                // tensor_dim0 hi16 (=0)
             | (((uint32_t)NH & 0xFFFFu) << 16));              // tensor_dim1 lo16 (=224)
    ld_g1[3] = (int32_t)((((uint32_t)NH) >> 16)                // tensor_dim1 hi16 (=0)
             | ((uint32_t)TUBE << 16));                        // tile_dim0 = 32
    ld_g1[4] = (int32_t)((uint32_t)TUBE                        // tile_dim1 = 32
             | ((uint32_t)NC << 16));                          // tile_dim2 = 3
    ld_g1[5] = (int32_t)NW;                                    // tensor_dim0_stride[31:0] = 224
    ld_g1[6] = (int32_t)(((uint32_t)plane & 0xFFFFu) << 16);   // d0s[47:32]=0 | dim1_stride lo16
    ld_g1[7] = (int32_t)((uint32_t)(plane >> 16));             // tensor_dim1_stride[47:16]

    i32x4 ld_g2;
    ld_g2[0] = NC;                                             // tensor_dim2 = 3
    ld_g2[1] = 0;                                              // tensor_dim3 (unused)
    ld_g2[2] = 0;                                              // tensor_dim2_stride lo (unused)
    ld_g2[3] = 0;                                              // stride hi | tile_dim3 = 0
    i32x4 ld_g3 = {0, 0, 0, 0};

    // ---------------- TDM store descriptor: 1D contiguous 3072 elements -------
    const uint32_t N = NC * TUBE * TUBE;                       // 3072
    u32x4 st_g0;
    st_g0[0] = 1u;
    st_g0[1] = lds_addr;
    st_g0[2] = (uint32_t)(dst_addr & 0xFFFFFFFFu);
    st_g0[3] = (uint32_t)((dst_addr >> 32) & 0x01FFFFFFu) | (2u << 30);

    i32x8 st_g1;
    st_g1[0] = (int32_t)(2u << 16);                            // data_size = 4 bytes
    st_g1[1] = (int32_t)((N & 0xFFFFu) << 16);                 // tensor_dim0 lo16 = 3072
    st_g1[2] = (int32_t)((N >> 16) | (1u << 16));              // tensor_dim0 hi | tensor_dim1=1
    st_g1[3] = (int32_t)(N << 16);                             // tile_dim0 = 3072
    st_g1[4] = 0;                                              // tile_dim1 = 0 (1D), tile_dim2 = 0
    st_g1[5] = (int32_t)N;                                     // tensor_dim0_stride
    st_g1[6] = 0;
    st_g1[7] = 0;
    i32x4 st_g2 = {0, 0, 0, 0};
    i32x4 st_g3 = {0, 0, 0, 0};

#if __clang_major__ >= 23
    i32x8 zpad = {0, 0, 0, 0, 0, 0, 0, 0};
    __builtin_amdgcn_tensor_load_to_lds(ld_g0, ld_g1, ld_g2, ld_g3, zpad, 0);
    __builtin_amdgcn_s_wait_tensorcnt(0);                      // LDS tile is complete
    __builtin_amdgcn_tensor_store_from_lds(st_g0, st_g1, st_g2, st_g3, zpad, 0);
#else
    __builtin_amdgcn_tensor_load_to_lds(ld_g0, ld_g1, ld_g2, ld_g3, 0);
    __builtin_amdgcn_s_wait_tensorcnt(0);                      // LDS tile is complete
    __builtin_amdgcn_tensor_store_from_lds(st_g0, st_g1, st_g2, st_g3, 0);
#endif
    // s_endpgm performs an implicit wait-idle, covering the outstanding store.
#else
    // Fallback (non-TDM toolchain): same geometry, plain gather copy.
    (void)lds_addr;
    for (uint32_t i = threadIdx.x; i < (uint32_t)(NC * TUBE * TUBE); i += 32u) {
        uint32_t c   = i / (TUBE * TUBE);
        uint32_t rem = i % (TUBE * TUBE);
        uint32_t row = rem / TUBE;
        uint32_t col = rem % TUBE;
        out[dst_elem + i] =
            video[src_elem + (uint64_t)c * plane + (uint64_t)row * NW + col];
    }
#endif
}

extern "C" void kernel_launch(void* const* d_in, const int* in_sizes, int n_in,
                              void* d_out, int out_size, void* d_ws, size_t ws_size,
                              hipStream_t stream) {
    const float* video = (const float*)d_in[0];
    const int*   yoff  = (const int*)d_in[1];
    const int*   xoff  = (const int*)d_in[2];
    float*       out   = (float*)d_out;

    // One wave32 workgroup per (b, tube, f) tube-frame: 8*8*16 = 1024 blocks.
    dim3 grid(NB * NTUBES * NF);
    dim3 block(32);
    VideoTube_tdm_kernel<<<grid, block, 0, stream>>>(video, yoff, xoff, out);
}